// ModelGRU_55765855371730
// MI455X (gfx1250) — compile-verified
//
#include <hip/hip_runtime.h>

typedef __attribute__((ext_vector_type(2))) float v2f;
typedef __attribute__((ext_vector_type(8))) float v8f;

#define B_ 8192
#define T_ 1024
#define I_ 3
#define H_ 4
#define G_ 12  // 3*H gates

// CDNA5 hardware TANH transcendental (v_tanh_f32), with exp/rcp fallback.
#if __has_builtin(__builtin_amdgcn_tanhf)
__device__ __forceinline__ float fast_tanh(float x) {
    return __builtin_amdgcn_tanhf(x);
}
#else
__device__ __forceinline__ float fast_tanh(float x) {
    return __builtin_amdgcn_rcpf(1.0f + __expf(-2.0f * x)) * 2.0f - 1.0f;
}
#endif

// One wave handles 16 batch rows. Lane pair (n, n+16) owns batch (base + n).
// Per timestep:
//   WMMA1: D_i[gate][batch] = A_i * x_t^T + C_i   (K = 3 input channels, padded to 4)
//   WMMA2: D_h[gate][batch] = A_h * h^T   + C_h   (K = 4 hidden)
// Gate rows 0-3 = r, 4-7 = z (land in lanes 0-15), 8-11 = n (land in lanes 16-31).
// Rows 0-7 of A_i/A_h/C_i/C_h are PRE-SCALED by 0.5 so the WMMA emits x/2 directly
// and sigmoid(x) = 0.5*tanh(x/2) + 0.5 needs only tanh + fma per gate.
// All stores are unconditional: after the bh rebuild, lower lanes hold (h0,h1) and
// upper lanes hold the true (h2,h3), so each lane stores its 8-byte half of out[b][t].
__global__ __launch_bounds__(256) void gru_wmma_kernel(
    const float* __restrict__ x,     // [B, T, I]
    const float* __restrict__ w_ih,  // [12, 3]
    const float* __restrict__ w_hh,  // [12, 4]
    const float* __restrict__ b_ih,  // [12]
    const float* __restrict__ b_hh,  // [12]
    float* __restrict__ out,         // [B, T, H]
    float* __restrict__ h_out)       // [B, H]
{
    const int lane = threadIdx.x & 31;
    const int wave = threadIdx.x >> 5;            // 0..7 waves per block
    const bool lo  = lane < 16;
    const int  m   = lane & 15;                   // A-row slot / batch slot in wave
    const int  b   = blockIdx.x * 128 + wave * 16 + m;  // batch owned by this lane pair

    // ---- A-matrix (16x4 f32) layout: lanes 0-15 -> K=0,1 ; lanes 16-31 -> K=2,3 ; M = lane&15
    const int   kb  = lo ? 0 : 2;
    const float asc = (m < 8) ? 0.5f : 1.0f;      // fold sigmoid's /2 into r/z gate rows
    v2f a_i, a_h;
    a_i.x = (m < G_ && kb     < I_) ? w_ih[m * I_ + kb]     * asc : 0.0f;
    a_i.y = (m < G_ && kb + 1 < I_) ? w_ih[m * I_ + kb + 1] * asc : 0.0f;  // K=3 col == 0
    a_h.x = (m < G_) ? w_hh[m * H_ + kb]     * asc : 0.0f;
    a_h.y = (m < G_) ? w_hh[m * H_ + kb + 1] * asc : 0.0f;

    // ---- Bias accumulators in C/D layout: lane VGPR r -> gate M = (lo ? r : 8+r)
    const int mb = lo ? 0 : 8;
    v8f ci, ch;
#pragma unroll
    for (int r = 0; r < 8; ++r) {
        const float bsc = (mb + r < 8) ? 0.5f : 1.0f;
        ci[r] = (mb + r < G_) ? b_ih[mb + r] * bsc : 0.0f;
        ch[r] = (mb + r < G_) ? b_hh[mb + r] * bsc : 0.0f;
    }

    // ---- Hidden state: owner (lower) lane keeps h[0..3]; bh holds h^T in B 4x16 layout
    float h0 = 0.0f, h1 = 0.0f, h2 = 0.0f, h3 = 0.0f;
    v2f bh;
    bh.x = 0.0f;   // K=0 (lo) / K=2 (hi), N = lane&15
    bh.y = 0.0f;   // K=1 (lo) / K=3 (hi)

    const float* xrow = x + (size_t)b * (T_ * I_);
    // Per-lane 8-byte slice of out[b][t][:] : lower -> h0,h1 ; upper -> h2,h3.
    float* ostore = out + (size_t)b * (T_ * H_) + (lo ? 0 : 2);

    for (int t = 0; t < T_; ++t) {
        // Branchless speculative prefetch ahead of the recurrence
        // (ISA: speculative prefetch with invalid address is silently dropped).
        __builtin_prefetch(xrow + (t + 8) * I_, 0, 3);

        // ---- B_x (x_t^T, 4x16, K = input channel)
        // bx.y in upper lanes feeds B row K=3, which multiplies the all-zero A column
        // K=3 — any finite value is fine, so no select is needed.
        float xv0 = xrow[t * I_ + (lo ? 0 : 2)];   // lo: x0 ; hi: x2  (always in bounds)
        float xv1 = xrow[t * I_ + 1];              // always in bounds
        v2f bx;
        bx.x = xv0;
        bx.y = xv1;

        v8f di = __builtin_amdgcn_wmma_f32_16x16x4_f32(
            false, a_i, false, bx, (short)0, ci, false, false);
        v8f dh = __builtin_amdgcn_wmma_f32_16x16x4_f32(
            false, a_h, false, bh, (short)0, ch, false, false);

        // lanes<16 : di[0..3]+dh[0..3] = (r pre-act)/2 ; di[4..7]+dh[4..7] = (z pre-act)/2
        // lanes>=16: di[0..3] = i_n, dh[0..3] = h_n (gates 8..11, unscaled); [4..7] = pad
        // sigmoid(x) = 0.5*tanh(x/2) + 0.5
        float r0 = 0.5f * fast_tanh(di[0] + dh[0]) + 0.5f;
        float r1 = 0.5f * fast_tanh(di[1] + dh[1]) + 0.5f;
        float r2 = 0.5f * fast_tanh(di[2] + dh[2]) + 0.5f;
        float r3 = 0.5f * fast_tanh(di[3] + dh[3]) + 0.5f;
        float z0 = 0.5f * fast_tanh(di[4] + dh[4]) + 0.5f;
        float z1 = 0.5f * fast_tanh(di[5] + dh[5]) + 0.5f;
        float z2 = 0.5f * fast_tanh(di[6] + dh[6]) + 0.5f;
        float z3 = 0.5f * fast_tanh(di[7] + dh[7]) + 0.5f;

        // Ship r (lower lanes) to upper lanes; compute candidate there.
        float rr0 = __shfl_xor(r0, 16, 32);
        float rr1 = __shfl_xor(r1, 16, 32);
        float rr2 = __shfl_xor(r2, 16, 32);
        float rr3 = __shfl_xor(r3, 16, 32);
        float c0 = fast_tanh(di[0] + rr0 * dh[0]);  // valid in upper lanes
        float c1 = fast_tanh(di[1] + rr1 * dh[1]);
        float c2 = fast_tanh(di[2] + rr2 * dh[2]);
        float c3 = fast_tanh(di[3] + rr3 * dh[3]);

        // Ship candidate back to owner (lower) lanes.
        float n0 = __shfl_xor(c0, 16, 32);
        float n1 = __shfl_xor(c1, 16, 32);
        float n2 = __shfl_xor(c2, 16, 32);
        float n3 = __shfl_xor(c3, 16, 32);

        // h_new = n + z*(h - n)  (meaningful in lower lanes; upper-lane values are
        // bounded garbage that is never consumed).
        h0 = n0 + z0 * (h0 - n0);
        h1 = n1 + z1 * (h1 - n1);
        h2 = n2 + z2 * (h2 - n2);
        h3 = n3 + z3 * (h3 - n3);

        // Rebuild bh (h^T in B 4x16 layout):
        //   VGPR0: K=0 in lanes 0-15 (h0), K=2 in lanes 16-31 (h2 from partner lane)
        //   VGPR1: K=1 in lanes 0-15 (h1), K=3 in lanes 16-31 (h3 from partner lane)
        float h2s = __shfl_xor(h2, 16, 32);
        float h3s = __shfl_xor(h3, 16, 32);
        bh.x = lo ? h0 : h2s;
        bh.y = lo ? h1 : h3s;

        // Unconditional 8B store: (bh.x, bh.y) is (h0,h1) in lower lanes and the
        // true (h2,h3) in upper lanes. 8-byte aligned.
        *(float2*)(ostore + t * H_) = make_float2(bh.x, bh.y);
    }

    // h_n [1, B, H]: same per-lane 8-byte halves, unconditional.
    *(float2*)(h_out + (size_t)b * H_ + (lo ? 0 : 2)) = make_float2(bh.x, bh.y);
}

extern "C" void kernel_launch(void* const* d_in, const int* in_sizes, int n_in,
                              void* d_out, int out_size, void* d_ws, size_t ws_size,
                              hipStream_t stream) {
    (void)in_sizes; (void)n_in; (void)out_size; (void)d_ws; (void)ws_size;
    const float* x    = (const float*)d_in[0];
    const float* w_ih = (const float*)d_in[1];
    const float* w_hh = (const float*)d_in[2];
    const float* b_ih = (const float*)d_in[3];
    const float* b_hh = (const float*)d_in[4];

    float* out   = (float*)d_out;                       // [B,T,H]
    float* h_out = out + (size_t)B_ * T_ * H_;          // [1,B,H] appended

    // 16 batches per wave, 8 waves per block -> 128 batches per block, 64 blocks.
    dim3 grid(B_ / 128);
    dim3 block(256);
    gru_wmma_kernel<<<grid, block, 0, stream>>>(x, w_ih, w_hh, b_ih, b_hh, out, h_out);
}